// Renderer_88880053224225
// MI455X (gfx1250) — compile-verified
//
#include <hip/hip_runtime.h>
#include <hip/hip_bf16.h>
#include <stdint.h>

#define N_POINTS 1000000
#define K_STEPS  32
#define N_FUNCS  8
#define HW_DIM   1024
#define HW_PIX   (HW_DIM * HW_DIM)
#define FID      1000

// ---------- helpers -------------------------------------------------------

// Zero a float4 region (used for the interleaved ws accumulator).
__global__ __launch_bounds__(256) void flame_zero4(float* __restrict__ dst, int n4) {
    int i = blockIdx.x * blockDim.x + threadIdx.x;
    if (i < n4) reinterpret_cast<float4*>(dst)[i] = make_float4(0.f, 0.f, 0.f, 0.f);
}

// Fallback path init: copy raw_image (zeros) into d_out.
__global__ __launch_bounds__(256) void flame_copy4(const float* __restrict__ src,
                                                   float* __restrict__ dst, int n4) {
    int i = blockIdx.x * blockDim.x + threadIdx.x;
    if (i < n4) reinterpret_cast<float4*>(dst)[i] =
                reinterpret_cast<const float4*>(src)[i];
}

// Interleaved path final: out[ch][p] = raw[ch][p] + ws[p*4+ch]  (p = x*W+y).
__global__ __launch_bounds__(256) void flame_combine(const float* __restrict__ raw,
                                                     const float* __restrict__ ws,
                                                     float* __restrict__ out) {
    int p = blockIdx.x * blockDim.x + threadIdx.x;
    if (p >= HW_PIX) return;
    float4 v = reinterpret_cast<const float4*>(ws)[p];
    out[0 * HW_PIX + p] = raw[0 * HW_PIX + p] + v.x;
    out[1 * HW_PIX + p] = raw[1 * HW_PIX + p] + v.y;
    out[2 * HW_PIX + p] = raw[2 * HW_PIX + p] + v.z;
    out[3 * HW_PIX + p] = raw[3 * HW_PIX + p] + v.w;
}

// ---------- main iteration kernel ----------------------------------------
// INTERLEAVED=true : acc is (H,W,4) -> 4 atomics per step share one 16B chunk.
// INTERLEAVED=false: acc is (4,H,W) -> direct accumulation into d_out.
template <bool INTERLEAVED>
__global__ __launch_bounds__(256) void flame_iter(
    const float* __restrict__ palette,   // (FID,4)
    const float* __restrict__ Amat,      // (8,2,2)
    const float* __restrict__ Bvec,      // (8,2)
    const float* __restrict__ fcolor,    // (8,)
    const float* __restrict__ xy0,       // (N,2)
    const float* __restrict__ c0,        // (N,1)
    const float* __restrict__ minv,      // (1,2)
    const float* __restrict__ rangev,    // (1,2)
    const int*   __restrict__ fn_idx,    // (K,N)
    const int*   __restrict__ skipk_p,   // scalar
    float*       __restrict__ acc)
{
    __shared__ float4 s_pal[FID];              // 16000 B
    __shared__ float  s_func[N_FUNCS * 8];     // a00 a01 a10 a11 bx by fc pad

    // --- Async DMA of the palette: global -> LDS, bypassing VGPRs (gfx1250) ---
    {
        unsigned ldsBase = (unsigned)(uintptr_t)(&s_pal[0]); // low 32b of flat = LDS offset
        for (int i = threadIdx.x; i < FID; i += 256) {
            unsigned     lds = ldsBase + i * 16u;
            const float* g   = palette + i * 4;
            asm volatile("global_load_async_to_lds_b128 %0, %1, off"
                         :: "v"(lds), "v"(g) : "memory");
        }
    }
    // --- Tiny function table into LDS (8 entries x 8 dwords across banks) ---
    if (threadIdx.x < N_FUNCS * 8) {
        int f = threadIdx.x >> 3, j = threadIdx.x & 7;
        float v = 0.0f;
        if (j < 4)       v = Amat[f * 4 + j];
        else if (j < 6)  v = Bvec[f * 2 + (j - 4)];
        else if (j == 6) v = fcolor[f];
        s_func[threadIdx.x] = v;
    }
    asm volatile("s_wait_asynccnt 0" ::: "memory");
    __syncthreads();

    const float minx = minv[0], miny = minv[1];
    const float rx   = rangev[0], ry = rangev[1];
    const int   skipk = skipk_p[0];

    const int n = blockIdx.x * 256 + threadIdx.x;
    if (n >= N_POINTS) return;

    const float2 xy = reinterpret_cast<const float2*>(xy0)[n];
    float x = xy.x, y = xy.y;
    float c = c0[n];

    // Issue all 32 coalesced fn_idx loads up front (rows are 4MB apart; this
    // maximizes memory-level parallelism; compiler inserts progressive
    // s_wait_loadcnt as the compute chain consumes them).
    int idxs[K_STEPS];
    #pragma unroll
    for (int k = 0; k < K_STEPS; ++k) idxs[k] = fn_idx[k * N_POINTS + n];

    #pragma unroll
    for (int k = 0; k < K_STEPS; ++k) {
        const float* f = &s_func[idxs[k] * 8];       // ds_load_b128 + ds_load_b96
        const float nx = f[0] * x + f[1] * y + f[4]; // einsum 'nij,nj->ni' + b
        const float ny = f[2] * x + f[3] * y + f[5];
        x = nx; y = ny;
        c = (c + f[6]) * 0.5f;

        int pi = (int)(c * (float)(FID - 1) + (0.5f / (float)FID)); // trunc == astype(int32)
        pi = pi < 0 ? 0 : (pi > FID - 1 ? FID - 1 : pi);
        const float4 col = s_pal[pi];                // ds_load_b128 gather

        const int xb = (int)((x - minx) * rx);       // trunc toward zero == jnp
        const int yb = (int)((y - miny) * ry);

        if (((unsigned)xb < (unsigned)HW_DIM) && ((unsigned)yb < (unsigned)HW_DIM)
            && (k >= skipk)) {
            const int p = xb * HW_DIM + yb;
            if (INTERLEAVED) {
                float* t = acc + (size_t)p * 4;      // one 16B chunk, one L2 line
                unsafeAtomicAdd(t + 0, col.x);
                unsafeAtomicAdd(t + 1, col.y);
                unsafeAtomicAdd(t + 2, col.z);
                unsafeAtomicAdd(t + 3, col.w);
            } else {
                unsafeAtomicAdd(acc + p,              col.x);
                unsafeAtomicAdd(acc + p + 1 * HW_PIX, col.y);
                unsafeAtomicAdd(acc + p + 2 * HW_PIX, col.z);
                unsafeAtomicAdd(acc + p + 3 * HW_PIX, col.w);
            }
        }
    }
}

// ---------- host launcher -------------------------------------------------

extern "C" void kernel_launch(void* const* d_in, const int* in_sizes, int n_in,
                              void* d_out, int out_size, void* d_ws, size_t ws_size,
                              hipStream_t stream) {
    const float* raw_image = (const float*)d_in[0];
    const float* palette   = (const float*)d_in[1];
    const float* Amat      = (const float*)d_in[2];
    const float* Bvec      = (const float*)d_in[3];
    const float* fcolor    = (const float*)d_in[4];
    const float* xy0       = (const float*)d_in[5];
    const float* c0        = (const float*)d_in[6];
    const float* minv      = (const float*)d_in[7];
    const float* rangev    = (const float*)d_in[8];
    const int*   fn_idx    = (const int*)d_in[9];
    const int*   skipk     = (const int*)d_in[10];
    float*       img       = (float*)d_out;

    const int    pixBlocks = (HW_PIX + 255) / 256;            // 4096
    const int    ptBlocks  = (N_POINTS + 255) / 256;          // 3907
    const size_t wsNeed    = (size_t)HW_PIX * 4 * sizeof(float); // 16 MB

    if (ws_size >= wsNeed && d_ws != nullptr) {
        // Interleaved (H,W,4) accumulator in ws: 4 atomics/step share a line.
        float* ws = (float*)d_ws;
        flame_zero4<<<pixBlocks, 256, 0, stream>>>(ws, HW_PIX);
        flame_iter<true><<<ptBlocks, 256, 0, stream>>>(
            palette, Amat, Bvec, fcolor, xy0, c0, minv, rangev, fn_idx, skipk, ws);
        flame_combine<<<pixBlocks, 256, 0, stream>>>(raw_image, ws, img);
    } else {
        // Fallback: accumulate directly into the (4,H,W) output.
        flame_copy4<<<pixBlocks, 256, 0, stream>>>(raw_image, img, HW_PIX);
        flame_iter<false><<<ptBlocks, 256, 0, stream>>>(
            palette, Amat, Bvec, fcolor, xy0, c0, minv, rangev, fn_idx, skipk, img);
    }
}